// RelativeMultiHeadAttention_89309549953500
// MI455X (gfx1250) — compile-verified
//
#include <hip/hip_runtime.h>
#include <math.h>

// ---------------- problem constants ----------------
#define DMODEL 512
#define NHEADS 8
#define DHEAD  64
#define SEQ    512
#define BATCH  8
#define MAXLEN 5000
#define NREL   1024                    // rel rows allocated (0..1022 used)
#define PE_ROW_OFF (MAXLEN - SEQ + 1)  // 4489: pe row = d' + 4489, d' = j-i+511

typedef float v2f __attribute__((ext_vector_type(2)));
typedef float v8f __attribute__((ext_vector_type(8)));

__device__ __forceinline__ v8f zero8() {
  v8f z = {0.f, 0.f, 0.f, 0.f, 0.f, 0.f, 0.f, 0.f};
  return z;
}

// Native fp32 tensor op: D(16x16) = A(16x4) * B(4x16) + C
__device__ __forceinline__ v8f wmma_f32(v2f a, v2f b, v8f c) {
  return __builtin_amdgcn_wmma_f32_16x16x4_f32(
      /*neg_a=*/false, a, /*neg_b=*/false, b,
      /*c_mod=*/(short)0, c, /*reuse_a=*/false, /*reuse_b=*/false);
}

// A fragment: row-major A, rows m0..m0+15, K = kb..kb+3
// lane l, reg r holds A[m0 + (l&15)][kb + 2*(l>>4) + r]  -> contiguous pair
__device__ __forceinline__ v2f fragA(const float* A, int lda, int m0, int kb, int lane) {
  const float* p = A + (size_t)(m0 + (lane & 15)) * lda + kb + 2 * (lane >> 4);
  float2 t = *(const float2*)p;
  v2f r; r.x = t.x; r.y = t.y;
  return r;
}
// Same addressing serves B[K][N] = W[N][K] for row-major W (transpose-free weight use).

// ---------------- kernel 1: rel[d'][n] = pe[d'+4489][:] . wr[n][:] ----------------
__global__ __launch_bounds__(256) void rel_proj_kernel(const float* __restrict__ pe,
                                                       const float* __restrict__ wr,
                                                       float* __restrict__ rel) {
  int lane = threadIdx.x & 31;
  int wid = blockIdx.x * (blockDim.x >> 5) + (threadIdx.x >> 5);
  if (wid >= (NREL / 16) * (DMODEL / 64)) return;     // 64*8 = 512 wave-tiles
  int mt = wid >> 3, nt = wid & 7;
  int m0 = mt * 16, n0 = nt * 64;
  const float* A = pe + (size_t)PE_ROW_OFF * DMODEL;

  v8f acc[4] = {zero8(), zero8(), zero8(), zero8()};
  for (int kb = 0; kb < DMODEL; kb += 4) {
    v2f a = fragA(A, DMODEL, m0, kb, lane);
#pragma unroll
    for (int s = 0; s < 4; ++s) {
      v2f b = fragA(wr, DMODEL, n0 + 16 * s, kb, lane);  // B[K][N] = wr[N][K]
      acc[s] = wmma_f32(a, b, acc[s]);
    }
  }
#pragma unroll
  for (int s = 0; s < 4; ++s)
#pragma unroll
    for (int r = 0; r < 8; ++r)
      rel[(size_t)(m0 + r + 8 * (lane >> 4)) * DMODEL + n0 + 16 * s + (lane & 15)] = acc[s][r];
}

// ---------------- kernel 2: Q/K/V projections, q fused with +u and +v_bias ----------------
__global__ __launch_bounds__(256) void qkv_kernel(const float* __restrict__ x,
                                                  const float* __restrict__ wq, const float* __restrict__ bq,
                                                  const float* __restrict__ wk, const float* __restrict__ bk,
                                                  const float* __restrict__ wv, const float* __restrict__ bv,
                                                  const float* __restrict__ u, const float* __restrict__ vbias,
                                                  float* __restrict__ qu, float* __restrict__ qv,
                                                  float* __restrict__ kmat, float* __restrict__ vmat) {
  int lane = threadIdx.x & 31;
  int wid = blockIdx.x * (blockDim.x >> 5) + (threadIdx.x >> 5);
  const int M = BATCH * SEQ;                 // 4096
  if (wid >= (M / 16) * (DMODEL / 64)) return;  // 2048 wave-tiles
  int mt = wid >> 3, nt = wid & 7;
  int m0 = mt * 16, n0 = nt * 64;
  int mat = blockIdx.y;
  const float* w    = (mat == 0) ? wq : (mat == 1) ? wk : wv;
  const float* bias = (mat == 0) ? bq : (mat == 1) ? bk : bv;

  v8f acc[4] = {zero8(), zero8(), zero8(), zero8()};
  for (int kb = 0; kb < DMODEL; kb += 4) {
    v2f a = fragA(x, DMODEL, m0, kb, lane);
#pragma unroll
    for (int s = 0; s < 4; ++s) {
      v2f b = fragA(w, DMODEL, n0 + 16 * s, kb, lane);
      acc[s] = wmma_f32(a, b, acc[s]);
    }
  }
#pragma unroll
  for (int s = 0; s < 4; ++s) {
    int n = n0 + 16 * s + (lane & 15);
    float bn = bias[n];
    int h = n >> 6, d = n & 63;
    float un = (mat == 0) ? u[n] : 0.f;
    float vn = (mat == 0) ? vbias[n] : 0.f;
#pragma unroll
    for (int r = 0; r < 8; ++r) {
      int m = m0 + r + 8 * (lane >> 4);
      int bidx = m >> 9, sidx = m & 511;
      size_t dst = (((size_t)bidx * NHEADS + h) * SEQ + sidx) * DHEAD + d;
      float val = acc[s][r] + bn;
      if (mat == 0) { qu[dst] = val + un; qv[dst] = val + vn; }
      else if (mat == 1) kmat[dst] = val;
      else               vmat[dst] = val;
    }
  }
}

// ---------------- kernel 3: fused relative attention per (b, h, 16-row block) ----------------
#define PSTR 536   // pscore strip stride (528 cols used)
#define SSTR 520   // score row stride (512 cols used), even -> 8B-aligned float2 LDS reads

__global__ __launch_bounds__(256) void attn_kernel(const float* __restrict__ qu,
                                                   const float* __restrict__ qv,
                                                   const float* __restrict__ kmat,
                                                   const float* __restrict__ vmat,
                                                   const float* __restrict__ rel,
                                                   float* __restrict__ attn_out) {
  __shared__ float qu_s[16 * 64];
  __shared__ float qv_s[16 * 64];
  __shared__ float p_s[16 * PSTR];
  __shared__ float s_s[16 * SSTR];
  __shared__ float o_s[16 * 64];

  int tid = threadIdx.x;
  int lane = tid & 31, wave = tid >> 5;
  int i0 = blockIdx.x * 16;
  int h  = blockIdx.y;
  int b  = blockIdx.z;
  size_t bh = ((size_t)b * NHEADS + h) * SEQ * DHEAD;

  // ---- stage A: tiles of (q+u), (q+v_bias) -> LDS; zero output accumulator ----
  {
    int row = tid >> 4, c4 = (tid & 15) * 4;
    *(float4*)&qu_s[row * 64 + c4] = *(const float4*)(qu + bh + (size_t)(i0 + row) * 64 + c4);
    *(float4*)&qv_s[row * 64 + c4] = *(const float4*)(qv + bh + (size_t)(i0 + row) * 64 + c4);
    *(float4*)&o_s[row * 64 + c4]  = make_float4(0.f, 0.f, 0.f, 0.f);
  }
  __syncthreads();

  // ---- stage B: position strip p[i][t] = qv_row(i) . rel[dbase+t][h*64:..], t in [0,528) ----
  int dbase = (SEQ - 16) - i0;   // 496 - i0  (>= 0; dbase+527 <= 1023 < NREL)
  for (int tt = wave; tt < 33; tt += 8) {
    int t0 = tt * 16;
    v8f acc = zero8();
    for (int kb = 0; kb < DHEAD; kb += 4) {
      v2f a;
      { const float* p = &qv_s[(lane & 15) * 64 + kb + 2 * (lane >> 4)]; a.x = p[0]; a.y = p[1]; }
      v2f bf;
      { const float* p = rel + (size_t)(dbase + t0 + (lane & 15)) * DMODEL + h * 64 + kb + 2 * (lane >> 4);
        float2 t = *(const float2*)p; bf.x = t.x; bf.y = t.y; }
      acc = wmma_f32(a, bf, acc);
    }
#pragma unroll
    for (int r = 0; r < 8; ++r)
      p_s[(r + 8 * (lane >> 4)) * PSTR + t0 + (lane & 15)] = acc[r];
  }
  __syncthreads();

  // ---- stage C: scores = (qu . k^T + position_gather) * scale ----
  const float scale = 0.125f;   // dh^-0.5
  for (int jt = wave; jt < 32; jt += 8) {
    int j0 = jt * 16;
    v8f acc = zero8();
    for (int kb = 0; kb < DHEAD; kb += 4) {
      v2f a;
      { const float* p = &qu_s[(lane & 15) * 64 + kb + 2 * (lane >> 4)]; a.x = p[0]; a.y = p[1]; }
      v2f bf;  // B[K=d][N=j] = k[j][d], contiguous in d
      { const float* p = kmat + bh + (size_t)(j0 + (lane & 15)) * 64 + kb + 2 * (lane >> 4);
        float2 t = *(const float2*)p; bf.x = t.x; bf.y = t.y; }
      acc = wmma_f32(a, bf, acc);
    }
#pragma unroll
    for (int r = 0; r < 8; ++r) {
      int il = r + 8 * (lane >> 4);
      int j  = j0 + (lane & 15);
      int t  = j - il + 15;                  // = (j - i_abs + 511) - dbase, in [0, 527)
      s_s[il * SSTR + j] = (acc[r] + p_s[il * PSTR + t]) * scale;
    }
  }
  __syncthreads();

  // ---- stage D: softmax over j (2 rows per wave, wave32 shuffles) ----
#pragma unroll
  for (int rr = 0; rr < 2; ++rr) {
    int row = wave * 2 + rr;
    float m = -3.402823466e38f;
    for (int c = lane; c < SEQ; c += 32) m = fmaxf(m, s_s[row * SSTR + c]);
#pragma unroll
    for (int off = 16; off; off >>= 1) m = fmaxf(m, __shfl_xor(m, off, 32));
    float sum = 0.f;
    for (int c = lane; c < SEQ; c += 32) {
      float e = __expf(s_s[row * SSTR + c] - m);
      s_s[row * SSTR + c] = e;
      sum += e;
    }
#pragma unroll
    for (int off = 16; off; off >>= 1) sum += __shfl_xor(sum, off, 32);
    float inv = 1.f / sum;
    for (int c = lane; c < SEQ; c += 32) s_s[row * SSTR + c] *= inv;
  }
  __syncthreads();

  // ---- stage E: out = attn @ V, K=512 split across 8 waves, LDS float-atomic reduce ----
  int kb0 = wave * 64;
#pragma unroll
  for (int nt = 0; nt < 4; ++nt) {
    int n0 = nt * 16;
    v8f acc = zero8();
    for (int kb = kb0; kb < kb0 + 64; kb += 4) {
      v2f a;
      { const float* p = &s_s[(lane & 15) * SSTR + kb + 2 * (lane >> 4)]; a.x = p[0]; a.y = p[1]; }
      v2f bf;  // B[K=j][N=d] = v[j][d], stride DHEAD between the two regs
      { const float* p = vmat + bh + (size_t)(kb + 2 * (lane >> 4)) * 64 + n0 + (lane & 15);
        bf.x = p[0]; bf.y = p[64]; }
      acc = wmma_f32(a, bf, acc);
    }
#pragma unroll
    for (int r = 0; r < 8; ++r)
      atomicAdd(&o_s[(r + 8 * (lane >> 4)) * 64 + n0 + (lane & 15)], acc[r]);
  }
  __syncthreads();

  // ---- stage F: write [B, S, H*dh] for the final projection ----
  {
    int row = tid >> 4, c4 = (tid & 15) * 4;
    *(float4*)(attn_out + ((size_t)b * SEQ + i0 + row) * DMODEL + h * 64 + c4) =
        *(float4*)&o_s[row * 64 + c4];
  }
}

// ---------------- kernel 4: out = attn_out @ wo^T + bo ----------------
__global__ __launch_bounds__(256) void oproj_kernel(const float* __restrict__ ain,
                                                    const float* __restrict__ wo,
                                                    const float* __restrict__ bo,
                                                    float* __restrict__ out) {
  int lane = threadIdx.x & 31;
  int wid = blockIdx.x * (blockDim.x >> 5) + (threadIdx.x >> 5);
  const int M = BATCH * SEQ;
  if (wid >= (M / 16) * (DMODEL / 64)) return;
  int mt = wid >> 3, nt = wid & 7;
  int m0 = mt * 16, n0 = nt * 64;

  v8f acc[4] = {zero8(), zero8(), zero8(), zero8()};
  for (int kb = 0; kb < DMODEL; kb += 4) {
    v2f a = fragA(ain, DMODEL, m0, kb, lane);
#pragma unroll
    for (int s = 0; s < 4; ++s) {
      v2f b = fragA(wo, DMODEL, n0 + 16 * s, kb, lane);
      acc[s] = wmma_f32(a, b, acc[s]);
    }
  }
#pragma unroll
  for (int s = 0; s < 4; ++s) {
    int n = n0 + 16 * s + (lane & 15);
    float bn = bo[n];
#pragma unroll
    for (int r = 0; r < 8; ++r) {
      int m = m0 + r + 8 * (lane >> 4);
      out[(size_t)m * DMODEL + n] = acc[s][r] + bn;
    }
  }
}

// ---------------- host launcher ----------------
extern "C" void kernel_launch(void* const* d_in, const int* in_sizes, int n_in,
                              void* d_out, int out_size, void* d_ws, size_t ws_size,
                              hipStream_t stream) {
  const float* x     = (const float*)d_in[0];
  const float* wq    = (const float*)d_in[1];
  const float* bq    = (const float*)d_in[2];
  const float* wk    = (const float*)d_in[3];
  const float* bk    = (const float*)d_in[4];
  const float* wv    = (const float*)d_in[5];
  const float* bv    = (const float*)d_in[6];
  const float* wo    = (const float*)d_in[7];
  const float* bo    = (const float*)d_in[8];
  const float* wr    = (const float*)d_in[9];
  const float* u     = (const float*)d_in[10];
  const float* vbias = (const float*)d_in[11];
  const float* pe    = (const float*)d_in[12];
  float* out = (float*)d_out;

  const size_t BHSD = (size_t)BATCH * NHEADS * SEQ * DHEAD;  // 2,097,152 floats
  float* ws   = (float*)d_ws;
  float* qu   = ws;
  float* qv   = qu + BHSD;
  float* kmat = qv + BHSD;
  float* vmat = kmat + BHSD;
  float* rel  = vmat + BHSD;                 // [NREL=1024, 512]
  float* ao   = rel + (size_t)NREL * DMODEL; // [B, S, D]
  // total: 44,040,192 bytes of d_ws

  // 1) relative-position projection: 1024x512x512 GEMM (512 wave-tiles -> 64 blocks)
  rel_proj_kernel<<<dim3(64), 256, 0, stream>>>(pe, wr, rel);

  // 2) Q/K/V projections: 4096x512x512 GEMM x3 (2048 wave-tiles -> 256 blocks each)
  qkv_kernel<<<dim3(256, 3), 256, 0, stream>>>(x, wq, bq, wk, bk, wv, bv, u, vbias,
                                               qu, qv, kmat, vmat);

  // 3) fused relative attention: one WG per (16-row block, head, batch)
  attn_kernel<<<dim3(SEQ / 16, NHEADS, BATCH), 256, 0, stream>>>(qu, qv, kmat, vmat, rel, ao);

  // 4) output projection + bias
  oproj_kernel<<<dim3(256), 256, 0, stream>>>(ao, wo, bo, out);
}